// LatentNode_88029649699591
// MI455X (gfx1250) — compile-verified
//
#include <hip/hip_runtime.h>
#include <hip/hip_bf16.h>

#define BB 64
#define SS 512
#define DMM 1024
#define DQQ 1024
#define DZZ 512
#define KK 1024

typedef __attribute__((ext_vector_type(2))) float v2f;
typedef __attribute__((ext_vector_type(8))) float v8f;

// ---------------------------------------------------------------------------
// WMMA f32 16x16x4 accumulation over a K range.
// C = A(M x K, row-major, lda) * Bt(N x K, row-major, ldb)^T
// A fragment per ISA: lane l holds A[l&15][k0 + 2*(l>>4) + {0,1}]
// B fragment per ISA: lane l holds B[k0 + 2*(l>>4) + {0,1}][l&15] = Bt[l&15][...]
// ---------------------------------------------------------------------------
__device__ inline v8f wmma_accum_f32(const float* __restrict__ A, int lda,
                                     const float* __restrict__ Bt, int ldb,
                                     int m0, int n0, int klen, v8f acc) {
    const int lane = threadIdx.x & 31;
    const int half = lane >> 4;
    const int l15  = lane & 15;
    const float* arow = A  + (size_t)(m0 + l15) * lda + 2 * half;
    const float* brow = Bt + (size_t)(n0 + l15) * ldb + 2 * half;
#pragma unroll 8
    for (int k = 0; k < klen; k += 4) {
        v2f a = *reinterpret_cast<const v2f*>(arow + k);
        v2f b = *reinterpret_cast<const v2f*>(brow + k);
        acc = __builtin_amdgcn_wmma_f32_16x16x4_f32(false, a, false, b,
                                                    (short)0, acc, false, false);
    }
    return acc;
}

// ---------------------------------------------------------------------------
// Kernel 1: q = init_query @ W_in^T    (64 x 1024, K = 1024)
// 1 wave per 16x16 tile; 4 x 64 tiles = 256 waves = 32 blocks x 8 waves
// ---------------------------------------------------------------------------
__global__ __launch_bounds__(256) void gemm_q_kernel(const float* __restrict__ iq,
                                                     const float* __restrict__ W_in,
                                                     float* __restrict__ q) {
    int wid = (blockIdx.x * blockDim.x + threadIdx.x) >> 5;
    int mt = wid & 3, nt = wid >> 2;           // mt<4, nt<64
    v8f acc = {};
    acc = wmma_accum_f32(iq, DQQ, W_in, DQQ, mt * 16, nt * 16, DQQ, acc);
    int lane = threadIdx.x & 31, half = lane >> 4, l15 = lane & 15;
    int n = nt * 16 + l15;
#pragma unroll
    for (int r = 0; r < 8; ++r) {
        int m = mt * 16 + r + 8 * half;
        q[(size_t)m * DMM + n] = acc[r];
    }
}

// ---------------------------------------------------------------------------
// Kernel 2: embnorm[k] = |emb[k]|^2   (1024 rows, DZ=512). One wave per row.
// ---------------------------------------------------------------------------
__global__ __launch_bounds__(256) void emb_norms_kernel(const float* __restrict__ emb,
                                                        float* __restrict__ en) {
    int row  = (blockIdx.x * blockDim.x + threadIdx.x) >> 5;
    int lane = threadIdx.x & 31;
    const float* r = emb + (size_t)row * DZZ;
    float s = 0.f;
#pragma unroll
    for (int d = lane; d < DZZ; d += 32) { float v = r[d]; s += v * v; }
    for (int m = 16; m; m >>= 1) s += __shfl_xor(s, m, 32);
    if (lane == 0) en[row] = s;
}

__global__ void init_keys_kernel(unsigned long long* __restrict__ keys) {
    if (threadIdx.x < BB) keys[threadIdx.x] = ~0ull;
}

// ---------------------------------------------------------------------------
// Kernel 3: masked-softmax attention, one block per batch, 512 threads.
//   scores[s] = q[b].mem[b,s,:]  (wave per row, float4 coalesced)
//   softmax with length mask, then ctx[d] = sum_s p[s]*mem[b,s,d]
// ---------------------------------------------------------------------------
__global__ __launch_bounds__(512) void attention_kernel(const float* __restrict__ mem,
                                                        const float* __restrict__ q,
                                                        const int* __restrict__ lens,
                                                        float* __restrict__ ctx) {
    __shared__ __align__(16) float qs[DMM];
    __shared__ float sc[SS];
    __shared__ float red[16];
    __shared__ float bcast;

    const int b = blockIdx.x;
    const int t = threadIdx.x;
    const int lane = t & 31, warp = t >> 5;           // 16 waves
    const float* qrow = q + (size_t)b * DMM;
    qs[t]       = qrow[t];
    qs[t + 512] = qrow[t + 512];
    __syncthreads();

    const int len = lens[b];
    const float* mb = mem + (size_t)b * SS * DMM;

    // -------- phase 1: scores (wave per row s) --------
    for (int it = 0; it < SS / 16; ++it) {
        int s = it * 16 + warp;
        const float* row = mb + (size_t)s * DMM;
        float acc = 0.f;
#pragma unroll
        for (int j = 0; j < 8; ++j) {
            int d = j * 128 + lane * 4;
            float4 mv = *reinterpret_cast<const float4*>(row + d);
            float4 qv = *reinterpret_cast<const float4*>(qs + d);
            acc += mv.x * qv.x + mv.y * qv.y + mv.z * qv.z + mv.w * qv.w;
        }
        for (int m = 16; m; m >>= 1) acc += __shfl_xor(acc, m, 32);
        if (lane == 0) sc[s] = (s < len) ? acc : -__builtin_inff();
    }
    __syncthreads();

    // -------- phase 2: softmax over sc[0..511] --------
    float v = sc[t];
    float mx = v;
    for (int m = 16; m; m >>= 1) mx = fmaxf(mx, __shfl_xor(mx, m, 32));
    if (lane == 0) red[warp] = mx;
    __syncthreads();
    if (t < 16) {
        float r = red[t];
        for (int m = 8; m; m >>= 1) r = fmaxf(r, __shfl_xor(r, m, 32));
        if (t == 0) bcast = r;
    }
    __syncthreads();
    float maxv = bcast;
    float e = __expf(v - maxv);
    float sm = e;
    for (int m = 16; m; m >>= 1) sm += __shfl_xor(sm, m, 32);
    __syncthreads();                  // everyone has read bcast(max); safe to reuse
    if (lane == 0) red[warp] = sm;
    __syncthreads();
    if (t < 16) {
        float r = red[t];
        for (int m = 8; m; m >>= 1) r += __shfl_xor(r, m, 32);
        if (t == 0) bcast = r;
    }
    __syncthreads();
    sc[t] = e * (1.0f / bcast);
    __syncthreads();

    // -------- phase 3: ctx (thread per 2 columns, coalesced rows) --------
    const int d0 = t * 2;
    float ax = 0.f, ay = 0.f;
    for (int s = 0; s < SS; ++s) {
        float w = sc[s];
        float2 mv = *reinterpret_cast<const float2*>(mb + (size_t)s * DMM + d0);
        ax += w * mv.x;
        ay += w * mv.y;
    }
    float* cr = ctx + (size_t)b * DMM + d0;
    cr[0] = ax;
    cr[1] = ay;
}

// ---------------------------------------------------------------------------
// Kernel 4: V = tanh([ctx, query] @ W_out^T)   (64 x 512, K = 2048)
// 4 x 32 tiles = 128 waves = 16 blocks x 8 waves
// ---------------------------------------------------------------------------
__global__ __launch_bounds__(256) void gemm_v_kernel(const float* __restrict__ ctx,
                                                     const float* __restrict__ iq,
                                                     const float* __restrict__ W_out,
                                                     float* __restrict__ V) {
    int wid = (blockIdx.x * blockDim.x + threadIdx.x) >> 5;
    int mt = wid & 3, nt = wid >> 2;            // mt<4, nt<32
    v8f acc = {};
    acc = wmma_accum_f32(ctx, DMM, W_out,       DQQ + DMM, mt * 16, nt * 16, DMM, acc);
    acc = wmma_accum_f32(iq,  DQQ, W_out + DMM, DQQ + DMM, mt * 16, nt * 16, DQQ, acc);
    int lane = threadIdx.x & 31, half = lane >> 4, l15 = lane & 15;
    int n = nt * 16 + l15;
#pragma unroll
    for (int r = 0; r < 8; ++r) {
        int m = mt * 16 + r + 8 * half;
        V[(size_t)m * DZZ + n] = tanhf(acc[r]);
    }
}

// ---------------------------------------------------------------------------
// Kernel 5: G = V @ emb^T, score = |emb_n|^2 - 2 G, packed-key argmin per batch
// 4 x 64 tiles = 256 waves = 32 blocks x 8 waves
// ---------------------------------------------------------------------------
__device__ inline unsigned ordered_bits(float f) {
    unsigned u = __float_as_uint(f);
    return (u & 0x80000000u) ? ~u : (u | 0x80000000u);
}

__global__ __launch_bounds__(256) void gemm_argmin_kernel(const float* __restrict__ V,
                                                          const float* __restrict__ emb,
                                                          const float* __restrict__ en,
                                                          unsigned long long* __restrict__ keys) {
    int wid = (blockIdx.x * blockDim.x + threadIdx.x) >> 5;
    int mt = wid & 3, nt = wid >> 2;            // mt<4, nt<64
    v8f acc = {};
    acc = wmma_accum_f32(V, DZZ, emb, DZZ, mt * 16, nt * 16, DZZ, acc);
    int lane = threadIdx.x & 31, half = lane >> 4, l15 = lane & 15;
    int n = nt * 16 + l15;
    float enorm = en[n];
#pragma unroll
    for (int r = 0; r < 8; ++r) {
        int m = mt * 16 + r + 8 * half;
        float score = enorm - 2.0f * acc[r];
        unsigned long long key =
            ((unsigned long long)ordered_bits(score) << 32) | (unsigned)n;
        // min over the 16 lanes sharing this half (xor masks keep bit 4 fixed)
        for (int msk = 1; msk < 16; msk <<= 1) {
            unsigned long long o = __shfl_xor(key, msk, 32);
            key = (o < key) ? o : key;
        }
        if (l15 == 0) atomicMin(&keys[m], key);
    }
}

// ---------------------------------------------------------------------------
// Kernel 6: finalize. collected[b] = (Wq + V) - V ; d1[b] = d2[b] = |Wq - V|^2
// ---------------------------------------------------------------------------
__global__ __launch_bounds__(256) void finalize_kernel(const float* __restrict__ V,
                                                       const float* __restrict__ emb,
                                                       const unsigned long long* __restrict__ keys,
                                                       float* __restrict__ out) {
    __shared__ float red[8];
    const int b = blockIdx.x;
    const int kmin = (int)(keys[b] & 0xFFFFFFFFull);
    const float* wq = emb + (size_t)kmin * DZZ;
    const float* vb = V + (size_t)b * DZZ;
    float local = 0.f;
    for (int d = threadIdx.x; d < DZZ; d += blockDim.x) {
        float w = wq[d], v = vb[d];
        out[(size_t)b * DZZ + d] = (w + v) - v;   // emulate sg(Wq)+V-sg(V) fp order
        float diff = w - v;
        local += diff * diff;
    }
    for (int m = 16; m; m >>= 1) local += __shfl_xor(local, m, 32);
    int warp = threadIdx.x >> 5, lane = threadIdx.x & 31;
    if (lane == 0) red[warp] = local;
    __syncthreads();
    if (warp == 0) {
        float r = (lane < 8) ? red[lane] : 0.f;
        for (int m = 4; m; m >>= 1) r += __shfl_xor(r, m, 32);
        if (lane == 0) {
            out[BB * DZZ + b]      = r;   // d1
            out[BB * DZZ + BB + b] = r;   // d2
        }
    }
}

// ---------------------------------------------------------------------------
extern "C" void kernel_launch(void* const* d_in, const int* in_sizes, int n_in,
                              void* d_out, int out_size, void* d_ws, size_t ws_size,
                              hipStream_t stream) {
    const float* mem   = (const float*)d_in[0];   // (B,S,DM)
    const float* iq    = (const float*)d_in[1];   // (B,DQ)
    const float* W_in  = (const float*)d_in[2];   // (DM,DQ)
    const float* W_out = (const float*)d_in[3];   // (DZ,DQ+DM)
    const float* emb   = (const float*)d_in[4];   // (K,DZ)
    const int*   lens  = (const int*)d_in[5];     // (B,)
    float* out = (float*)d_out;

    float* ws  = (float*)d_ws;
    float* q   = ws;                 // B*DM   = 65536 floats
    float* ctx = q + BB * DMM;       // B*DM   = 65536 floats
    float* V   = ctx + BB * DMM;     // B*DZ   = 32768 floats
    float* en  = V + BB * DZZ;       // K      = 1024 floats
    unsigned long long* keys = (unsigned long long*)(en + KK);  // B u64

    gemm_q_kernel   <<<32, 256, 0, stream>>>(iq, W_in, q);
    emb_norms_kernel<<<KK / 8, 256, 0, stream>>>(emb, en);
    init_keys_kernel<<<1, 64, 0, stream>>>(keys);
    attention_kernel<<<BB, 512, 0, stream>>>(mem, q, lens, ctx);
    gemm_v_kernel   <<<16, 256, 0, stream>>>(ctx, iq, W_out, V);
    gemm_argmin_kernel<<<32, 256, 0, stream>>>(V, emb, en, keys);
    finalize_kernel <<<BB, 256, 0, stream>>>(V, emb, keys, out);
}